// DifferentiableCensus_27522150432942
// MI455X (gfx1250) — compile-verified
//
#include <hip/hip_runtime.h>

#define IMG      512
#define TILE     64
#define HALO     66      // TILE + 2 (1-pixel clamped halo)
#define LSTRIDE  68      // padded LDS row stride (floats)
#define NTHREADS 256     // 8 waves (wave32); 16x16 threads, each owns 4x4 pixels

// ---- CDNA5 feature gates (device pass only; host pass takes fallbacks) ----
#if defined(__gfx1250__) && __has_builtin(__builtin_amdgcn_global_load_async_to_lds_b32)
#define USE_ASYNC_LDS 1
#else
#define USE_ASYNC_LDS 0
#endif

#if defined(__gfx1250__) && __has_builtin(__builtin_amdgcn_tanhf)
#define USE_HW_TANH 1
#else
#define USE_HW_TANH 0
#endif

typedef __attribute__((address_space(1))) int gbl_int_t;   // global (AS1) int
typedef __attribute__((address_space(3))) int lds_int_t;   // LDS (AS3) int

// tanh(d/2): one V_TANH_F32 on gfx1250; fallback = 2*sigmoid(d)-1 via v_exp_f32+v_rcp_f32
__device__ __forceinline__ float tanh_half(float d) {
#if USE_HW_TANH
  return __builtin_amdgcn_tanhf(0.5f * d);
#else
  float e = __builtin_amdgcn_exp2f(-1.44269504088896340736f * d);
  return __builtin_amdgcn_rcpf(1.0f + e) * 2.0f - 1.0f;
#endif
}

__global__ __launch_bounds__(NTHREADS) void census_tanh_kernel(
    const float* __restrict__ x, float* __restrict__ out) {
  __shared__ float tile[HALO][LSTRIDE];

  const int tid  = threadIdx.x;
  const int row0 = blockIdx.y * TILE;
  const int col0 = blockIdx.x * TILE;
  const long base = (long)blockIdx.z * (IMG * IMG);
  const float* __restrict__ src = x + base;

  // ---- Stage 66x66 clamped region into LDS (clamp == edge padding) ----
  for (int idx = tid; idx < HALO * HALO; idx += NTHREADS) {
    const int r = idx / HALO;
    const int c = idx - r * HALO;
    int gr = row0 - 1 + r; gr = gr < 0 ? 0 : (gr > IMG - 1 ? IMG - 1 : gr);
    int gc = col0 - 1 + c; gc = gc < 0 ? 0 : (gc > IMG - 1 ? IMG - 1 : gc);
#if USE_ASYNC_LDS
    __builtin_amdgcn_global_load_async_to_lds_b32(
        (gbl_int_t*)(src + (gr * IMG + gc)),
        (lds_int_t*)&tile[r][c],
        0 /*imm offset*/, 0 /*cpol*/);
#else
    tile[r][c] = src[gr * IMG + gc];
#endif
  }
#if USE_ASYNC_LDS
#if __has_builtin(__builtin_amdgcn_s_wait_asynccnt)
  __builtin_amdgcn_s_wait_asynccnt(0);
#else
  asm volatile("s_wait_asynccnt 0" ::: "memory");
#endif
#endif
  __syncthreads();

  // ---- Each thread: 4x4 output block, reads its 6x6 LDS neighborhood ----
  const int tx = tid & 15;
  const int ty = tid >> 4;
  const int lr = ty * 4;   // block-local pixel row of this thread's 4x4 block
  const int lc = tx * 4;

  // L[i+1][j+1] = x(i,j) for i,j in [-1,4] relative to the 4x4 block origin
  float L[6][6];
#pragma unroll
  for (int r = 0; r < 6; ++r)
#pragma unroll
    for (int c = 0; c < 6; ++c)
      L[r][c] = tile[lr + r][lc + c];

  float acc[4][4];
#pragma unroll
  for (int i = 0; i < 4; ++i)
#pragma unroll
    for (int j = 0; j < 4; ++j) acc[i][j] = 0.0f;

  // tanh is odd: each pair edge evaluated once, +t to one pixel, -t to the other.
  // Horizontal edges: t = tanh(0.5*(x[i][j+1]-x[i][j])), i in [0,3], j in [-1,3]
#pragma unroll
  for (int i = 0; i < 4; ++i)
#pragma unroll
    for (int j = -1; j <= 3; ++j) {
      const float h = tanh_half(L[i + 1][j + 2] - L[i + 1][j + 1]);
      if (j >= 0)     acc[i][j]     += h;
      if (j + 1 <= 3) acc[i][j + 1] -= h;
    }
  // Vertical edges: t = tanh(0.5*(x[i+1][j]-x[i][j])), i in [-1,3], j in [0,3]
#pragma unroll
  for (int i = -1; i <= 3; ++i)
#pragma unroll
    for (int j = 0; j < 4; ++j) {
      const float v = tanh_half(L[i + 2][j + 1] - L[i + 1][j + 1]);
      if (i >= 0)     acc[i][j]     += v;
      if (i + 1 <= 3) acc[i + 1][j] -= v;
    }
  // Diagonal down-right edges: t = tanh(0.5*(x[i+1][j+1]-x[i][j]))
#pragma unroll
  for (int i = -1; i <= 3; ++i)
#pragma unroll
    for (int j = -1; j <= 3; ++j) {
      if ((i == -1 && j == 3) || (i == 3 && j == -1)) continue;  // unused corners
      const float d = tanh_half(L[i + 2][j + 2] - L[i + 1][j + 1]);
      if (i >= 0 && j >= 0)         acc[i][j]         += d;
      if (i + 1 <= 3 && j + 1 <= 3) acc[i + 1][j + 1] -= d;
    }
  // Diagonal down-left edges: t = tanh(0.5*(x[i+1][j-1]-x[i][j])), j in [0,4]
#pragma unroll
  for (int i = -1; i <= 3; ++i)
#pragma unroll
    for (int j = 0; j <= 4; ++j) {
      if ((i == -1 && j == 0) || (i == 3 && j == 4)) continue;   // unused corners
      const float d = tanh_half(L[i + 2][j] - L[i + 1][j + 1]);
      if (i >= 0 && j <= 3)         acc[i][j]         += d;
      if (i + 1 <= 3 && j - 1 >= 0) acc[i + 1][j - 1] -= d;
    }

  // out = (Σ9 sigmoid)/9 = 0.5 + (Σ8 tanh(d/2))/18 ; center term is tanh(0)=0
  float* __restrict__ dst = out + base + (long)(row0 + lr) * IMG + (col0 + lc);
#pragma unroll
  for (int i = 0; i < 4; ++i) {
    float4 v;
    v.x = fmaf(acc[i][0], 1.0f / 18.0f, 0.5f);
    v.y = fmaf(acc[i][1], 1.0f / 18.0f, 0.5f);
    v.z = fmaf(acc[i][2], 1.0f / 18.0f, 0.5f);
    v.w = fmaf(acc[i][3], 1.0f / 18.0f, 0.5f);
    *reinterpret_cast<float4*>(dst + (long)i * IMG) = v;  // 16B-aligned
  }
}

extern "C" void kernel_launch(void* const* d_in, const int* in_sizes, int n_in,
                              void* d_out, int out_size, void* d_ws, size_t ws_size,
                              hipStream_t stream) {
  (void)n_in; (void)out_size; (void)d_ws; (void)ws_size;
  const float* x = (const float*)d_in[0];
  float* out = (float*)d_out;
  const int planes = in_sizes[0] / (IMG * IMG);  // B*C = 48 for (16,3,512,512)
  dim3 grid(IMG / TILE, IMG / TILE, planes);     // 8 x 8 x 48 = 3072 workgroups
  census_tanh_kernel<<<grid, dim3(NTHREADS), 0, stream>>>(x, out);
}